// Pacemaker_49349174231265
// MI455X (gfx1250) — compile-verified
//
#include <hip/hip_runtime.h>
#include <hip/hip_bf16.h>

// ---------------------------------------------------------------------------
// Pacemaker OU + Bernoulli spikes for MI455X (gfx1250, wave32).
//
//   x_t = a*x_{t-1} + s*xi_t,  a = 1 - theta*dt,  s = sigma*sqrt(dt)
//   spike[t][n] = (u_{t,n} < clip(rate*dt + x_t, 0, 1))
//
// Strategy (affine-scan decomposition, 3 kernels):
//   pass1: per (slice, neuron) accumulate c_s = sum_j a^{S-j} n_j   (RNG only)
//   pass2: per neuron, 64-long serial scan -> x_start[slice][n]
//   pass3: per (slice, 16-neuron group) wave: 16-step chunks via
//          V_WMMA_F32_16X16X4_F32:  X = D*Noise + a^{r+1} (x) x0
//          then Bernoulli + non-temporal streaming store of 512MB output.
// ---------------------------------------------------------------------------

typedef __attribute__((ext_vector_type(2))) float v2f;
typedef __attribute__((ext_vector_type(8))) float v8f;

#define N_NEURONS 8192
#define T_STEPS   16384
#define SLICE     256                    // timesteps per slice
#define NSLICES   (T_STEPS / SLICE)      // 64
#define CHUNK     16
#define NCHUNKS   (SLICE / CHUNK)        // 16
#define NGROUPS   (N_NEURONS / 16)       // 512 neuron groups of 16

#define DECAY_A   0.999f                 // 1 - theta*dt
#define NSCALE    (0.6f * 0.03162277660168379f)  // sigma*sqrt(dt)
#define PBASE     0.003f                 // rate*dt

// --- stateless counter-based RNG (same bits in pass1 and pass3) -------------
__device__ __forceinline__ unsigned int pm_pcg(unsigned int x) {
  x = x * 747796405u + 2891336453u;
  unsigned int w = ((x >> ((x >> 28) + 4u)) ^ x) * 277803737u;
  return (w >> 22) ^ w;
}
__device__ __forceinline__ float pm_u01(unsigned int h) {
  // (0,1], 24-bit mantissa
  return (float)((h >> 8) + 1u) * 5.960464477539063e-8f;
}
// Box-Muller pair for timesteps (t_even, t_even+1) of neuron n.
__device__ __forceinline__ void pm_normal_pair(unsigned int t_even, unsigned int n,
                                               float* z0, float* z1) {
  unsigned int key = t_even * (unsigned int)N_NEURONS + n;   // < 2^27, no overflow
  float u1 = pm_u01(pm_pcg(key ^ 0x9E3779B9u));
  float u2 = pm_u01(pm_pcg(key ^ 0x85EBCA6Bu));
  float r  = __fsqrt_rn(-2.0f * __logf(u1));
  float a  = 6.28318530718f * u2;
  *z0 = r * __cosf(a);
  *z1 = r * __sinf(a);
}
__device__ __forceinline__ float pm_bern_u(unsigned int t, unsigned int n) {
  unsigned int key = t * (unsigned int)N_NEURONS + n;
  return pm_u01(pm_pcg(key ^ 0xC2B2AE35u));
}

// --- pass 1: per-slice noise accumulation c_s[n] ----------------------------
__global__ void pm_pass1_contrib(float* __restrict__ cbuf) {
  int gid   = blockIdx.x * blockDim.x + threadIdx.x;   // 0 .. NSLICES*N-1
  int slice = gid / N_NEURONS;
  int n     = gid - slice * N_NEURONS;
  unsigned int t0 = (unsigned int)(slice * SLICE);
  float c = 0.0f;
  for (int m = 0; m < SLICE; m += 2) {
    float z0, z1;
    pm_normal_pair(t0 + (unsigned int)m, (unsigned int)n, &z0, &z1);
    c = __builtin_fmaf(DECAY_A, c, z0 * NSCALE);
    c = __builtin_fmaf(DECAY_A, c, z1 * NSCALE);
  }
  cbuf[slice * N_NEURONS + n] = c;   // c = sum_j a^{S-j} n_j
}

// --- pass 2: tiny cross-slice scan (64 steps per neuron) --------------------
__global__ void pm_pass2_scan(const float* __restrict__ state0,
                              const float* __restrict__ cbuf,
                              float* __restrict__ xstart) {
  int n = blockIdx.x * blockDim.x + threadIdx.x;
  float aS = DECAY_A;                 // a^256 via 8 squarings
#pragma unroll
  for (int i = 0; i < 8; ++i) aS *= aS;
  float x = state0[n];
  for (int s = 0; s < NSLICES; ++s) {
    xstart[s * N_NEURONS + n] = x;
    x = __builtin_fmaf(aS, x, cbuf[s * N_NEURONS + n]);
  }
}

// --- pass 3: WMMA chunk scan + spike emission -------------------------------
__global__ void pm_pass3_emit(const float* __restrict__ xstart,
                              float* __restrict__ out) {
  const int lane  = threadIdx.x & 31;
  const int wid   = blockIdx.x * (blockDim.x >> 5) + (threadIdx.x >> 5);
  const int group = wid & (NGROUPS - 1);     // 512 groups of 16 neurons
  const int slice = wid >> 9;                // 64 slices
  const int n0    = group * 16;
  const int r     = lane & 15;               // A: row M ; B/C/D: neuron column
  const int hi    = lane >> 4;

  // Constant A-matrix D[r][c] = (c<=r) ? a^{r-c} : 0, split over 4 K=4 WMMAs.
  // Lane layout: WMMA i, VGPR0 -> column 4i + 2*hi, VGPR1 -> +1.
  float Ae[8];
#pragma unroll
  for (int i = 0; i < 8; ++i) Ae[i] = 0.0f;
  {
    float pw = 1.0f;
#pragma unroll
    for (int j = 0; j < 16; ++j) {
#pragma unroll
      for (int i = 0; i < 4; ++i) {
        int c0 = 4 * i + 2 * hi;
        if (r - c0 == j)     Ae[2 * i]     = pw;
        if (r - c0 - 1 == j) Ae[2 * i + 1] = pw;
      }
      pw *= DECAY_A;
    }
  }
  // x0 carry-in coefficients: C/D VGPR v holds row M = v + 8*hi -> a^{M+1}.
  float coef[8];
  {
    float base = DECAY_A;                          // a^1
    if (hi) {
      float a2 = DECAY_A * DECAY_A, a4 = a2 * a2, a8 = a4 * a4;
      base = a8 * DECAY_A;                         // a^9
    }
    coef[0] = base;
#pragma unroll
    for (int v = 1; v < 8; ++v) coef[v] = coef[v - 1] * DECAY_A;
  }

  float x0 = xstart[slice * N_NEURONS + n0 + r];   // both half-waves load same x0
  const int tslice = slice * SLICE;

  for (int ch = 0; ch < NCHUNKS; ++ch) {
    const int tb = tslice + ch * CHUNK;
    v8f acc = {};
#pragma unroll
    for (int i = 0; i < 4; ++i) {
      // B VGPR0 -> noise row k = 4i+2hi (t = tb+k), VGPR1 -> k+1 : one BM pair.
      float z0, z1;
      pm_normal_pair((unsigned int)(tb + 4 * i + 2 * hi), (unsigned int)(n0 + r),
                     &z0, &z1);
      v2f Bv; Bv.x = z0 * NSCALE; Bv.y = z1 * NSCALE;
      v2f Av; Av.x = Ae[2 * i];   Av.y = Ae[2 * i + 1];
      acc = __builtin_amdgcn_wmma_f32_16x16x4_f32(
          /*neg_a=*/false, Av, /*neg_b=*/false, Bv,
          /*c_mod=*/(short)0, acc, /*reuse_a=*/false, /*reuse_b=*/false);
    }
    // Add carry-in term, emit spikes, stream to HBM with NT hint.
    float xlast = 0.0f;
#pragma unroll
    for (int v = 0; v < 8; ++v) {
      float x  = __builtin_fmaf(coef[v], x0, acc[v]);
      int   t  = tb + v + hi * 8;
      int   nn = n0 + r;
      float p  = fminf(fmaxf(PBASE + x, 0.0f), 1.0f);
      float sp = (pm_bern_u((unsigned int)t, (unsigned int)nn) < p) ? 1.0f : 0.0f;
      __builtin_nontemporal_store(sp, out + (size_t)t * N_NEURONS + nn);
      if (v == 7) xlast = x;
    }
    // Row 15 lives in VGPR7 of lanes 16..31; broadcast to the low half-wave.
    x0 = __shfl(xlast, (lane & 15) + 16, 32);
  }
}

// ---------------------------------------------------------------------------
extern "C" void kernel_launch(void* const* d_in, const int* in_sizes, int n_in,
                              void* d_out, int out_size, void* d_ws, size_t ws_size,
                              hipStream_t stream) {
  (void)in_sizes; (void)n_in; (void)out_size; (void)ws_size;
  const float* state0 = (const float*)d_in[0];   // initial OU state (zeros)
  float* out   = (float*)d_out;                  // [T, N] f32 spikes
  float* cbuf  = (float*)d_ws;                                   // 2 MB
  float* xst   = cbuf + (size_t)NSLICES * N_NEURONS;             // 2 MB

  pm_pass1_contrib<<<(NSLICES * N_NEURONS) / 256, 256, 0, stream>>>(cbuf);
  pm_pass2_scan  <<<N_NEURONS / 256,              256, 0, stream>>>(state0, cbuf, xst);
  pm_pass3_emit  <<<(NSLICES * NGROUPS) / 8,      256, 0, stream>>>(xst, out);
}